// GATBlock_30279519437601
// MI455X (gfx1250) — compile-verified
//
#include <hip/hip_runtime.h>
#include <math.h>

// Problem constants (fixed by the reference)
#define B_    4
#define N_    2048
#define E_    65536
#define DIN   33
#define HID   32
#define HEADS 4
#define OUTC  8

typedef float v2f __attribute__((ext_vector_type(2)));
typedef float v8f __attribute__((ext_vector_type(8)));

// V_WMMA_F32_16X16X4_F32: exact f32 semantics, 8-arg builtin pattern
__device__ __forceinline__ v8f wmma_f32(v2f a, v2f b, v8f c) {
  return __builtin_amdgcn_wmma_f32_16x16x4_f32(false, a, false, b, (short)0, c,
                                               false, false);
}

__device__ __forceinline__ float leaky(float x) { return x > 0.f ? x : 0.2f * x; }

// float atomic max via int/uint trick (valid with -inf init)
__device__ __forceinline__ void atomicMaxF(float* addr, float v) {
  if (v >= 0.f) atomicMax((int*)addr, __float_as_int(v));
  else          atomicMin((unsigned int*)addr, __float_as_uint(v));
}

// ---------------------------------------------------------------------------
// Kernel 1: fused MLP + LayerNorm + GAT projection + attention coefficients.
// One wave32 per 16-node tile; WMMA f32 16x16x4 for all three GEMMs.
// ---------------------------------------------------------------------------
__global__ __launch_bounds__(128) void k_embed(
    const float* __restrict__ H,  const float* __restrict__ w1, const float* __restrict__ b1,
    const float* __restrict__ w2, const float* __restrict__ b2,
    const float* __restrict__ lng, const float* __restrict__ lnb,
    const float* __restrict__ gw, const float* __restrict__ attS, const float* __restrict__ attD,
    float* __restrict__ xh, float* __restrict__ asrc, float* __restrict__ adst)
{
  __shared__ float W1s[36 * 32];   // w1^T, K padded 33->36
  __shared__ float W2s[32 * 32];   // w2^T
  __shared__ float GWs[32 * 32];   // gat_w^T
  __shared__ float Tile[4][16 * 37];

  const int tid = threadIdx.x;
  for (int i = tid; i < 36 * 32; i += 128) {
    int k = i >> 5, n = i & 31;
    W1s[i] = (k < DIN) ? w1[n * DIN + k] : 0.f;
  }
  for (int i = tid; i < 32 * 32; i += 128) {
    int k = i >> 5, n = i & 31;
    W2s[i] = w2[n * 32 + k];
    GWs[i] = gw[n * 32 + k];
  }
  __syncthreads();

  const int wave = tid >> 5, lane = tid & 31;
  const int gnode0 = (blockIdx.x * 4 + wave) * 16;  // flattened b*N+n, tile base
  float* T = &Tile[wave][0];
  const int TS = 37;

  // Stage H tile [16 x 36], zero-padded K
  for (int i = lane; i < 16 * 36; i += 32) {
    int r = i / 36, c = i - r * 36;
    T[r * TS + c] = (c < DIN) ? H[(size_t)(gnode0 + r) * DIN + c] : 0.f;
  }

  const int m   = lane & 15;           // A-fragment row
  const int klo = (lane >> 4) << 1;    // 0 or 2 : K sub-offset for this half-wave
  const int n16 = lane & 15;           // B/C/D column within 16-wide tile
  const int mb  = (lane >> 4) << 3;    // 0 or 8 : C/D row base

  // ---- GEMM1: x1 = relu(H @ w1^T + b1)  (K = 36) ----
  v8f acc0 = {}, acc1 = {};
  #pragma unroll
  for (int kk = 0; kk < 9; ++kk) {
    int k = kk * 4 + klo;
    v2f av;  av.x  = T[m * TS + k];            av.y  = T[m * TS + k + 1];
    v2f bv0; bv0.x = W1s[k * 32 + n16];        bv0.y = W1s[(k + 1) * 32 + n16];
    v2f bv1; bv1.x = W1s[k * 32 + 16 + n16];   bv1.y = W1s[(k + 1) * 32 + 16 + n16];
    acc0 = wmma_f32(av, bv0, acc0);
    acc1 = wmma_f32(av, bv1, acc1);
  }
  {
    float bb0 = b1[n16], bb1 = b1[16 + n16];
    #pragma unroll
    for (int i = 0; i < 8; ++i) {
      T[(mb + i) * TS + n16]      = fmaxf(acc0[i] + bb0, 0.f);
      T[(mb + i) * TS + 16 + n16] = fmaxf(acc1[i] + bb1, 0.f);
    }
  }

  // ---- GEMM2: x2 = relu(x1 @ w2^T + b2)  (K = 32) ----
  v8f d0 = {}, d1 = {};
  #pragma unroll
  for (int kk = 0; kk < 8; ++kk) {
    int k = kk * 4 + klo;
    v2f av;  av.x  = T[m * TS + k];            av.y  = T[m * TS + k + 1];
    v2f bv0; bv0.x = W2s[k * 32 + n16];        bv0.y = W2s[(k + 1) * 32 + n16];
    v2f bv1; bv1.x = W2s[k * 32 + 16 + n16];   bv1.y = W2s[(k + 1) * 32 + 16 + n16];
    d0 = wmma_f32(av, bv0, d0);
    d1 = wmma_f32(av, bv1, d1);
  }
  {
    float bb0 = b2[n16], bb1 = b2[16 + n16];
    #pragma unroll
    for (int i = 0; i < 8; ++i) {
      T[(mb + i) * TS + n16]      = fmaxf(d0[i] + bb0, 0.f);
      T[(mb + i) * TS + 16 + n16] = fmaxf(d1[i] + bb1, 0.f);
    }
  }

  // ---- LayerNorm over 32 channels (one node per lane 0..15) ----
  if (lane < 16) {
    float mu = 0.f;
    #pragma unroll
    for (int c = 0; c < 32; ++c) mu += T[lane * TS + c];
    mu *= (1.f / 32.f);
    float var = 0.f;
    #pragma unroll
    for (int c = 0; c < 32; ++c) { float dd = T[lane * TS + c] - mu; var += dd * dd; }
    var *= (1.f / 32.f);
    float rs = rsqrtf(var + 1e-5f);
    #pragma unroll
    for (int c = 0; c < 32; ++c)
      T[lane * TS + c] = (T[lane * TS + c] - mu) * rs * lng[c] + lnb[c];
  }

  // ---- GEMM3: xh = x @ gat_w^T  (K = 32, no bias here) ----
  v8f g0 = {}, g1 = {};
  #pragma unroll
  for (int kk = 0; kk < 8; ++kk) {
    int k = kk * 4 + klo;
    v2f av;  av.x  = T[m * TS + k];            av.y  = T[m * TS + k + 1];
    v2f bv0; bv0.x = GWs[k * 32 + n16];        bv0.y = GWs[(k + 1) * 32 + n16];
    v2f bv1; bv1.x = GWs[k * 32 + 16 + n16];   bv1.y = GWs[(k + 1) * 32 + 16 + n16];
    g0 = wmma_f32(av, bv0, g0);
    g1 = wmma_f32(av, bv1, g1);
  }
  #pragma unroll
  for (int i = 0; i < 8; ++i) {
    float v0i = g0[i], v1i = g1[i];
    xh[(size_t)(gnode0 + mb + i) * HID + n16]      = v0i;
    xh[(size_t)(gnode0 + mb + i) * HID + 16 + n16] = v1i;
    T[(mb + i) * TS + n16]      = v0i;
    T[(mb + i) * TS + 16 + n16] = v1i;
  }

  // ---- attention coefficients a_src / a_dst ----
  if (lane < 16) {
    int node = gnode0 + lane;
    #pragma unroll
    for (int h = 0; h < HEADS; ++h) {
      float s = 0.f, d = 0.f;
      #pragma unroll
      for (int c = 0; c < OUTC; ++c) {
        float v = T[lane * TS + h * OUTC + c];
        s += v * attS[h * OUTC + c];
        d += v * attD[h * OUTC + c];
      }
      asrc[(size_t)node * HEADS + h] = s;
      adst[(size_t)node * HEADS + h] = d;
    }
  }
}

// ---------------------------------------------------------------------------
// Init / zero kernels
// ---------------------------------------------------------------------------
__global__ void k_init_small(float* __restrict__ out, float* __restrict__ mbuf,
                             float* __restrict__ dnm, const float* __restrict__ bias) {
  int i = blockIdx.x * blockDim.x + threadIdx.x;
  if (i < B_ * N_ * HID) out[i] = bias[i & 31];
  if (i < N_ * B_ * HEADS) { mbuf[i] = -__builtin_inff(); dnm[i] = 0.f; }
}

__global__ void k_zeroA(float4* __restrict__ A4) {
  size_t i = (size_t)blockIdx.x * blockDim.x + threadIdx.x;
  A4[i] = make_float4(0.f, 0.f, 0.f, 0.f);
}

// ---------------------------------------------------------------------------
// Edge passes: segment max -> segment sum of exp -> scatter (out, A)
// One thread per (edge, batch, head); logits recomputed each pass (cheap).
// ---------------------------------------------------------------------------
__global__ void k_edge_max(const int* __restrict__ ei, const float* __restrict__ asrc,
                           const float* __restrict__ adst, float* __restrict__ mbuf) {
  int idx = blockIdx.x * blockDim.x + threadIdx.x;     // E*B*H threads
  int e = idx >> 4, r = idx & 15, b = r >> 2, h = r & 3;
  int s = ei[e], d = ei[E_ + e];
  float lg = leaky(asrc[(size_t)(b * N_ + s) * HEADS + h] +
                   adst[(size_t)(b * N_ + d) * HEADS + h]);
  atomicMaxF(&mbuf[((size_t)d * B_ + b) * HEADS + h], lg);
}

__global__ void k_edge_sum(const int* __restrict__ ei, const float* __restrict__ asrc,
                           const float* __restrict__ adst, const float* __restrict__ mbuf,
                           float* __restrict__ dnm) {
  int idx = blockIdx.x * blockDim.x + threadIdx.x;
  int e = idx >> 4, r = idx & 15, b = r >> 2, h = r & 3;
  int s = ei[e], d = ei[E_ + e];
  size_t key = ((size_t)d * B_ + b) * HEADS + h;
  float lg = leaky(asrc[(size_t)(b * N_ + s) * HEADS + h] +
                   adst[(size_t)(b * N_ + d) * HEADS + h]);
  atomicAdd(&dnm[key], __expf(lg - mbuf[key]));
}

__global__ void k_edge_scatter(const int* __restrict__ ei, const float* __restrict__ asrc,
                               const float* __restrict__ adst, const float* __restrict__ mbuf,
                               const float* __restrict__ dnm, const float* __restrict__ xh,
                               float* __restrict__ out, float* __restrict__ A) {
  int idx = blockIdx.x * blockDim.x + threadIdx.x;
  int e = idx >> 4, r = idx & 15, b = r >> 2, h = r & 3;
  int s = ei[e], d = ei[E_ + e];
  size_t key = ((size_t)d * B_ + b) * HEADS + h;
  float lg = leaky(asrc[(size_t)(b * N_ + s) * HEADS + h] +
                   adst[(size_t)(b * N_ + d) * HEADS + h]);
  float alpha = __expf(lg - mbuf[key]) / (dnm[key] + 1e-16f);

  const float4* xs = (const float4*)(xh + (size_t)(b * N_ + s) * HID + h * OUTC);
  float4 p0 = xs[0], p1 = xs[1];
  float* od = out + (size_t)(b * N_ + d) * HID + h * OUTC;
  atomicAdd(&od[0], alpha * p0.x); atomicAdd(&od[1], alpha * p0.y);
  atomicAdd(&od[2], alpha * p0.z); atomicAdd(&od[3], alpha * p0.w);
  atomicAdd(&od[4], alpha * p1.x); atomicAdd(&od[5], alpha * p1.y);
  atomicAdd(&od[6], alpha * p1.z); atomicAdd(&od[7], alpha * p1.w);

  atomicAdd(&A[(((size_t)(b * HEADS + h)) * N_ + s) * N_ + d], alpha);
}

// ---------------------------------------------------------------------------
// Row-normalize A over dst axis: one 256-thread block per (b,h,src) row.
// ---------------------------------------------------------------------------
__global__ __launch_bounds__(256) void k_rownorm(float* __restrict__ A) {
  __shared__ float red[256];
  float* row = A + (size_t)blockIdx.x * N_;
  float s = 0.f;
  #pragma unroll
  for (int i = 0; i < N_ / 256; ++i) s += row[threadIdx.x + i * 256];
  red[threadIdx.x] = s;
  __syncthreads();
  for (int off = 128; off > 0; off >>= 1) {
    if (threadIdx.x < off) red[threadIdx.x] += red[threadIdx.x + off];
    __syncthreads();
  }
  float inv = 1.f / fmaxf(red[0], 1e-9f);
  #pragma unroll
  for (int i = 0; i < N_ / 256; ++i) row[threadIdx.x + i * 256] *= inv;
}

// ---------------------------------------------------------------------------
extern "C" void kernel_launch(void* const* d_in, const int* in_sizes, int n_in,
                              void* d_out, int out_size, void* d_ws, size_t ws_size,
                              hipStream_t stream) {
  const float* H   = (const float*)d_in[0];
  const int*   ei  = (const int*)d_in[1];   // [2,E] int32
  const float* w1  = (const float*)d_in[2];
  const float* b1  = (const float*)d_in[3];
  const float* w2  = (const float*)d_in[4];
  const float* b2  = (const float*)d_in[5];
  const float* lng = (const float*)d_in[6];
  const float* lnb = (const float*)d_in[7];
  const float* gw  = (const float*)d_in[8];
  const float* aS  = (const float*)d_in[9];
  const float* aD  = (const float*)d_in[10];
  const float* gb  = (const float*)d_in[11];

  float* out = (float*)d_out;                    // [B,N,32]
  float* A   = out + (size_t)B_ * N_ * HID;      // [B,H,N,N]

  // Workspace layout (~1.5 MB floats)
  float* ws   = (float*)d_ws;
  float* xh   = ws;                                   // B*N*32
  float* asrc = xh   + (size_t)B_ * N_ * HID;         // B*N*H
  float* adst = asrc + (size_t)B_ * N_ * HEADS;       // B*N*H
  float* mbuf = adst + (size_t)B_ * N_ * HEADS;       // N*B*H
  float* dnm  = mbuf + (size_t)N_ * B_ * HEADS;       // N*B*H

  k_init_small<<<(B_ * N_ * HID + 255) / 256, 256, 0, stream>>>(out, mbuf, dnm, gb);
  k_zeroA<<<(B_ * HEADS * (size_t)N_ * N_ / 4 + 255) / 256, 256, 0, stream>>>((float4*)A);
  k_embed<<<(B_ * N_) / 64, 128, 0, stream>>>(H, w1, b1, w2, b2, lng, lnb, gw, aS, aD,
                                              xh, asrc, adst);
  k_edge_max<<<(E_ * B_ * HEADS) / 256, 256, 0, stream>>>(ei, asrc, adst, mbuf);
  k_edge_sum<<<(E_ * B_ * HEADS) / 256, 256, 0, stream>>>(ei, asrc, adst, mbuf, dnm);
  k_edge_scatter<<<(E_ * B_ * HEADS) / 256, 256, 0, stream>>>(ei, asrc, adst, mbuf, dnm,
                                                              xh, out, A);
  k_rownorm<<<B_ * HEADS * N_, 256, 0, stream>>>(A);
}